// Reorg_18975165513946
// MI455X (gfx1250) — compile-verified
//
#include <hip/hip_runtime.h>

// Reorg / space-to-depth, stride 2, on x: (B=8, C=64, H=512, W=512) fp32.
//   out[b, (sh*2+sw)*C + c, h', w'] = x[b, c, 2h'+sh, 2w'+sw]
//
// Pure data movement: 512 MiB in + 512 MiB out = 1 GiB traffic.
// MI455X HBM = 23.3 TB/s  ->  ~46 us floor. Zero FLOPs -> WMMA inapplicable;
// TDM/LDS staging unnecessary since both loads and stores are fully
// coalescible directly. Strategy: per-thread 32B load (2x b128), split
// even/odd lanes of W, 2x contiguous b128 stores, non-temporal hints
// (streams >> 192MB L2, data touched exactly once).

typedef __attribute__((ext_vector_type(4))) float v4f;

#define B_  8u
#define C_  64u
#define H_  512u
#define W_  512u
// derived (stride 2)
#define H2_ 256u
#define W2_ 256u
// elements per (output) channel plane and per-C block
#define OUT_CBLOCK (C_ * H2_ * W2_)   // 4,194,304 : distance from sw=0 to sw=1 block

__global__ __launch_bounds__(256) void reorg_s2d_kernel(
    const float* __restrict__ in, float* __restrict__ out) {
  // One thread per 8 consecutive input floats along W.
  // idx bit layout (powers of two): [5:0]=w8, [14:6]=h, [20:15]=c, [23:21]=b
  const unsigned idx = blockIdx.x * 256u + threadIdx.x;   // 2^24 threads total

  const unsigned w8 = idx & 63u;            // which 8-float group along W
  const unsigned h  = (idx >> 6) & 511u;
  const unsigned c  = (idx >> 15) & 63u;
  const unsigned b  = idx >> 21;

  const unsigned hp = h >> 1;               // h'
  const unsigned sh = h & 1u;

  // Input is linear in (b,c,h,w8) exactly like idx: offset = idx * 8 floats.
  // 32-byte aligned -> two b128 loads.
  const v4f* src = reinterpret_cast<const v4f*>(in) + (idx << 1);
  const v4f v0 = __builtin_nontemporal_load(src);       // w = 8w8 .. 8w8+3
  const v4f v1 = __builtin_nontemporal_load(src + 1);   // w = 8w8+4 .. 8w8+7

  // Even-w elements -> channel (sh*2+0)*C + c ; odd-w -> +C block.
  // Deinterleave: even w' = 4w8..4w8+3 contiguous.
  const v4f evn = { v0.x, v0.z, v1.x, v1.z };
  const v4f odd = { v0.y, v0.w, v1.y, v1.w };

  const unsigned chanE = (sh << 7) + c;                       // (sh*2)*64 + c
  const unsigned baseE = ((((b << 8) + chanE) << 8) | hp) << 8;
  const unsigned wq    = w8 << 2;                             // w' start

  v4f* dstE = reinterpret_cast<v4f*>(out + (baseE + wq));
  v4f* dstO = reinterpret_cast<v4f*>(out + (baseE + wq + OUT_CBLOCK));
  __builtin_nontemporal_store(evn, dstE);
  __builtin_nontemporal_store(odd, dstO);
}

extern "C" void kernel_launch(void* const* d_in, const int* in_sizes, int n_in,
                              void* d_out, int out_size, void* d_ws, size_t ws_size,
                              hipStream_t stream) {
  (void)in_sizes; (void)n_in; (void)out_size; (void)d_ws; (void)ws_size;
  const float* x = (const float*)d_in[0];
  float* out = (float*)d_out;

  const unsigned total_threads = (B_ * C_ * H_ * W_) / 8u;  // 16,777,216
  dim3 block(256);
  dim3 grid(total_threads / 256u);                          // 65,536 blocks
  reorg_s2d_kernel<<<grid, block, 0, stream>>>(x, out);
}